// FusionLoss_7206955122711
// MI455X (gfx1250) — compile-verified
//
#include <hip/hip_runtime.h>
#include <math.h>

// ---------------------------------------------------------------------------
// FusionLoss for MI455X / gfx1250 (wave32, WMMA).
// Shapes fixed by reference: B=16, H=W=512, 3 channels, f32.
// ---------------------------------------------------------------------------

typedef __attribute__((ext_vector_type(2))) float v2f;
typedef __attribute__((ext_vector_type(8))) float v8f;

#define BN    16
#define HH    512
#define WW    512
#define CHW   (HH * WW)        // 262144
#define NPIX  (BN * CHW)       // 4194304
#define SSIM_C1 0.0001f
#define SSIM_C2 0.0009f

// ---------------------------------------------------------------------------
// f32 WMMA: D(16x16) = A(16x4) * B(4x16) + C   (V_WMMA_F32_16X16X4_F32)
// A: lane<16 -> row=lane, K={0,1}; lane>=16 -> row=lane-16, K={2,3}
// B: lane<16 -> K={0,1}, col=lane; lane>=16 -> K={2,3}, col=lane-16
// C/D: VGPR v -> row v (lanes 0-15) / row v+8 (lanes 16-31), col = lane&15
// ---------------------------------------------------------------------------
__device__ __forceinline__ v8f wmma_f32_16x16x4(v2f a, v2f b, v8f c) {
  return __builtin_amdgcn_wmma_f32_16x16x4_f32(
      /*neg_a=*/false, a, /*neg_b=*/false, b,
      /*c_mod=*/(short)0, c, /*reuse_a=*/false, /*reuse_b=*/false);
}

// ===========================================================================
// Kernel 1: denorm + clip + RGB->Y for the 3 images; MSE(ir) block partials.
// ===========================================================================
__global__ __launch_bounds__(256) void prep_y_mse(
    const float* __restrict__ fused, const float* __restrict__ vis,
    const float* __restrict__ ir,
    const float* __restrict__ mean_vis, const float* __restrict__ std_vis,
    const float* __restrict__ mean_ir,  const float* __restrict__ std_ir,
    float* __restrict__ Yf, float* __restrict__ Yv, float* __restrict__ Yi,
    float* __restrict__ msePart) {
  const int idx = blockIdx.x * 256 + threadIdx.x;   // pixel in [0, NPIX)
  const int b   = idx >> 18;                        // / CHW
  const int p   = idx & (CHW - 1);
  const size_t base = (size_t)b * 3 * CHW + p;

  const float wY[3] = {0.299f, 0.587f, 0.114f};
  float yf = 0.f, yv = 0.f, yi = 0.f;
#pragma unroll
  for (int c = 0; c < 3; ++c) {
    const float mv = mean_vis[c], sv = std_vis[c];
    const float mi = mean_ir[c],  si = std_ir[c];
    float f = fmaf(fused[base + (size_t)c * CHW], sv, mv);
    float v = fmaf(vis  [base + (size_t)c * CHW], sv, mv);
    float r = fmaf(ir   [base + (size_t)c * CHW], si, mi);
    f = fminf(fmaxf(f, 0.f), 1.f);
    v = fminf(fmaxf(v, 0.f), 1.f);
    r = fminf(fmaxf(r, 0.f), 1.f);
    yf = fmaf(wY[c], f, yf);
    yv = fmaf(wY[c], v, yv);
    yi = fmaf(wY[c], r, yi);
  }
  Yf[idx] = yf; Yv[idx] = yv; Yi[idx] = yi;

  const float d = yf - yi;
  __shared__ float red[256];
  red[threadIdx.x] = d * d;
  __syncthreads();
#pragma unroll
  for (int s = 128; s > 0; s >>= 1) {
    if (threadIdx.x < s) red[threadIdx.x] += red[threadIdx.x + s];
    __syncthreads();
  }
  if (threadIdx.x == 0) msePart[blockIdx.x] = red[0];
}

// ===========================================================================
// Kernel 2: SSIM(Yf,Yv) via separable 11-tap Gaussian as banded-matrix WMMAs.
// One 16x16 output tile per wave; 4 waves per block.
//   vertical:   V(16x32) = Gv(16x28 band) * X(28x32)   -> 2x7 WMMAs / map
//   horizontal: O(16x16) = V(16x28)       * Gh(28x16)  ->   7 WMMAs / map
// 5 maps (x, y, x^2, y^2, xy) -> 105 v_wmma_f32_16x16x4_f32 per tile.
// ===========================================================================
__global__ __launch_bounds__(128) void ssim_wmma(
    const float* __restrict__ Yf, const float* __restrict__ Yv,
    float* __restrict__ ssimPart) {
  __shared__ float sXf[4][28 * 32];   // padded halo tile of Yf (rows 26,27 = 0)
  __shared__ float sXv[4][28 * 32];   // padded halo tile of Yv
  __shared__ float sV [4][16 * 33];   // vertical-pass spill (row stride 33)
  __shared__ float sG [48];           // zero-padded Gaussian band: sG[16+i]=g[i]
  __shared__ float sRed[4];

  const int wave = threadIdx.x >> 5;
  const int lane = threadIdx.x & 31;
  const int half = lane >> 4;
  const int n    = lane & 15;

  // Gaussian taps (f32, same formula as reference), zero padded to [-16, 31].
  if (threadIdx.x < 48) {
    const int j = (int)threadIdx.x - 16;
    float val = 0.f;
    if (j >= 0 && j <= 10) {
      float norm = 0.f;
#pragma unroll
      for (int i = 0; i < 11; ++i) {
        const float c = (float)i - 5.f;
        norm += expf(-(c * c) / 4.5f);          // 2*sigma^2 = 4.5
      }
      const float cj = (float)j - 5.f;
      val = expf(-(cj * cj) / 4.5f) / norm;
    }
    sG[threadIdx.x] = val;
  }

  // Tile coordinates: 32x32 tiles of 16x16 per image.
  const int tile = blockIdx.x * 4 + wave;
  const int b    = tile >> 10;
  const int rem  = tile & 1023;
  const int r0   = (rem >> 5) << 4;
  const int c0   = (rem & 31) << 4;

  const float* pf = Yf + (size_t)b * CHW;
  const float* pv = Yv + (size_t)b * CHW;

  // Stage 26x32 halo (zero padded; conv uses explicit zero padding), +2 zero rows.
#pragma unroll 4
  for (int u = 0; u < 28; ++u) {
    const int gr = r0 - 5 + u;
    const int gc = c0 - 5 + lane;
    float vf = 0.f, vv = 0.f;
    if (u < 26 && gr >= 0 && gr < HH && gc >= 0 && gc < WW) {
      vf = pf[gr * WW + gc];
      vv = pv[gr * WW + gc];
    }
    sXf[wave][u * 32 + lane] = vf;
    sXv[wave][u * 32 + lane] = vv;
  }
  __syncthreads();

  v8f conv[5];
#pragma unroll
  for (int m = 0; m < 5; ++m) {
    // ---- vertical pass: A = Gaussian band, B = map values from LDS ----
    v8f v0 = {};                       // V cols 0..15
    v8f v1 = {};                       // V cols 16..31
#pragma unroll
    for (int kc = 0; kc < 7; ++kc) {
      const int u0 = kc * 4 + half * 2;       // K rows handled by this lane
      v2f a;                                   // Gv[row=n][u0], Gv[n][u0+1]
      a.x = sG[16 + u0 - n];
      a.y = sG[16 + u0 + 1 - n];
      const float f00 = sXf[wave][u0 * 32 + n];
      const float x00 = sXv[wave][u0 * 32 + n];
      const float f10 = sXf[wave][(u0 + 1) * 32 + n];
      const float x10 = sXv[wave][(u0 + 1) * 32 + n];
      const float f01 = sXf[wave][u0 * 32 + 16 + n];
      const float x01 = sXv[wave][u0 * 32 + 16 + n];
      const float f11 = sXf[wave][(u0 + 1) * 32 + 16 + n];
      const float x11 = sXv[wave][(u0 + 1) * 32 + 16 + n];
      v2f b0, b1;
      b0.x = (m == 0) ? f00 : (m == 1) ? x00 : (m == 2) ? f00 * f00
                            : (m == 3) ? x00 * x00 : f00 * x00;
      b0.y = (m == 0) ? f10 : (m == 1) ? x10 : (m == 2) ? f10 * f10
                            : (m == 3) ? x10 * x10 : f10 * x10;
      b1.x = (m == 0) ? f01 : (m == 1) ? x01 : (m == 2) ? f01 * f01
                            : (m == 3) ? x01 * x01 : f01 * x01;
      b1.y = (m == 0) ? f11 : (m == 1) ? x11 : (m == 2) ? f11 * f11
                            : (m == 3) ? x11 * x11 : f11 * x11;
      v0 = wmma_f32_16x16x4(a, b0, v0);
      v1 = wmma_f32_16x16x4(a, b1, v1);
    }

    // Spill V (C-layout) to LDS so it can be re-read in A-layout.
#pragma unroll
    for (int v = 0; v < 8; ++v) {
      sV[wave][(v + half * 8) * 33 + n]      = v0[v];
      sV[wave][(v + half * 8) * 33 + 16 + n] = v1[v];
    }
    __syncthreads();

    // ---- horizontal pass: A = V rows, B = Gaussian band ----
    v8f h = {};
#pragma unroll
    for (int kc = 0; kc < 7; ++kc) {
      const int k0 = kc * 4 + half * 2;
      v2f a;                                   // V[row=n][k0], V[n][k0+1]
      a.x = sV[wave][n * 33 + k0];
      a.y = sV[wave][n * 33 + k0 + 1];
      v2f bb;                                  // Gh[k0][col=n], Gh[k0+1][n]
      bb.x = sG[16 + k0 - n];
      bb.y = sG[16 + k0 + 1 - n];
      h = wmma_f32_16x16x4(a, bb, h);
    }
    conv[m] = h;
    __syncthreads();                           // WAR on sV before next map
  }

  // SSIM map over this lane's 8 elements of the 16x16 tile.
  float ts = 0.f;
#pragma unroll
  for (int v = 0; v < 8; ++v) {
    const float mux = conv[0][v], muy = conv[1][v];
    const float cxx = conv[2][v], cyy = conv[3][v], cxy = conv[4][v];
    const float mux2 = mux * mux, muy2 = muy * muy, muxy = mux * muy;
    const float sx2 = cxx - mux2, sy2 = cyy - muy2, sxy = cxy - muxy;
    const float num = (2.f * muxy + SSIM_C1) * (2.f * sxy + SSIM_C2);
    const float den = (mux2 + muy2 + SSIM_C1) * (sx2 + sy2 + SSIM_C2);
    ts += num / den;
  }
#pragma unroll
  for (int off = 16; off > 0; off >>= 1) ts += __shfl_down(ts, off, 32);
  if (lane == 0) sRed[wave] = ts;
  __syncthreads();
  if (threadIdx.x == 0)
    ssimPart[blockIdx.x] = sRed[0] + sRed[1] + sRed[2] + sRed[3];
}

// ===========================================================================
// Kernel 3: Sobel magnitudes + 6 absolute-difference block partial sums.
// ===========================================================================
__device__ __forceinline__ float ldpad(const float* s, int r, int c) {
  return (r >= 0 && r < HH && c >= 0 && c < WW) ? s[r * WW + c] : 0.f;
}

__global__ __launch_bounds__(256) void sobel_kernel(
    const float* __restrict__ Yf, const float* __restrict__ Yv,
    const float* __restrict__ Yi, float* __restrict__ sobPart, int nBlocks) {
  const int idx = blockIdx.x * 256 + threadIdx.x;
  const int b = idx >> 18;
  const int p = idx & (CHW - 1);
  const int r = p >> 9;
  const int c = p & 511;
  const size_t base = (size_t)b * CHW;
  const float* planes[3] = {Yf + base, Yv + base, Yi + base};

  float gx[3], gy[3];
#pragma unroll
  for (int q = 0; q < 3; ++q) {
    const float* s = planes[q];
    const float a00 = ldpad(s, r - 1, c - 1), a01 = ldpad(s, r - 1, c),
                a02 = ldpad(s, r - 1, c + 1);
    const float a10 = ldpad(s, r, c - 1), a12 = ldpad(s, r, c + 1);
    const float a20 = ldpad(s, r + 1, c - 1), a21 = ldpad(s, r + 1, c),
                a22 = ldpad(s, r + 1, c + 1);
    gx[q] = fabsf((a02 - a00) + 2.f * (a12 - a10) + (a22 - a20));
    gy[q] = fabsf((a00 + 2.f * a01 + a02) - (a20 + 2.f * a21 + a22));
  }
  float t[6];
  t[0] = fabsf(gx[0] - gx[1]);
  t[1] = fabsf(gy[0] - gy[1]);
  t[2] = fabsf(gx[0] - gx[2]);
  t[3] = fabsf(gy[0] - gy[2]);
  t[4] = fabsf(gx[0] - fmaxf(gx[1], gx[2]));
  t[5] = fabsf(gy[0] - fmaxf(gy[1], gy[2]));

  __shared__ float red[256];
#pragma unroll
  for (int k = 0; k < 6; ++k) {
    red[threadIdx.x] = t[k];
    __syncthreads();
#pragma unroll
    for (int s = 128; s > 0; s >>= 1) {
      if (threadIdx.x < s) red[threadIdx.x] += red[threadIdx.x + s];
      __syncthreads();
    }
    if (threadIdx.x == 0) sobPart[k * nBlocks + blockIdx.x] = red[0];
    __syncthreads();
  }
}

// ===========================================================================
// Kernel 4: deterministic final reduction + weighted combination -> scalar.
// ===========================================================================
__global__ __launch_bounds__(256) void finalize_kernel(
    const float* __restrict__ msePart, int nMse,
    const float* __restrict__ ssimPart, int nSsim,
    const float* __restrict__ sobPart, int nSob, float* __restrict__ out) {
  __shared__ float red[256];
  const int tid = threadIdx.x;
  float sums[8];
  const float* arrs[8];
  int lens[8];
  arrs[0] = msePart;  lens[0] = nMse;
  arrs[1] = ssimPart; lens[1] = nSsim;
#pragma unroll
  for (int k = 0; k < 6; ++k) { arrs[2 + k] = sobPart + (size_t)k * nSob; lens[2 + k] = nSob; }

  for (int k = 0; k < 8; ++k) {
    float acc = 0.f;
    for (int i = tid; i < lens[k]; i += 256) acc += arrs[k][i];
    red[tid] = acc;
    __syncthreads();
#pragma unroll
    for (int s = 128; s > 0; s >>= 1) {
      if (tid < s) red[tid] += red[tid + s];
      __syncthreads();
    }
    sums[k] = red[0];
    __syncthreads();
  }
  if (tid == 0) {
    const float inv = 1.f / (float)NPIX;
    const float loss_mse  = sums[0] * inv;                               // W_MSE_IR=1
    const float loss_ssim = 1.f - sums[1] * inv;                         // W_SSIM_VIS=1
    const float loss_sob  = (sums[2] + sums[3] + sums[4] + sums[5]) * inv;
    const float loss_grad = (sums[6] + sums[7]) * inv;                   // W_GRAD=1
    out[0] = loss_ssim + loss_sob + loss_grad + loss_mse;
  }
}

// ===========================================================================
// Launch
// ===========================================================================
extern "C" void kernel_launch(void* const* d_in, const int* in_sizes, int n_in,
                              void* d_out, int out_size, void* d_ws, size_t ws_size,
                              hipStream_t stream) {
  const float* fused    = (const float*)d_in[0];
  const float* vis      = (const float*)d_in[1];
  const float* ir       = (const float*)d_in[2];
  const float* mean_vis = (const float*)d_in[3];
  const float* std_vis  = (const float*)d_in[4];
  const float* mean_ir  = (const float*)d_in[5];
  const float* std_ir   = (const float*)d_in[6];
  (void)in_sizes; (void)n_in; (void)out_size; (void)ws_size;

  float* ws = (float*)d_ws;
  float* Yf = ws;
  float* Yv = Yf + NPIX;
  float* Yi = Yv + NPIX;
  const int nMse  = NPIX / 256;        // 16384 blocks
  const int nSsim = (BN * 32 * 32) / 4; // 16384 tiles, 4 per block -> 4096
  const int nSob  = NPIX / 256;        // 16384 blocks
  float* msePart  = Yi + NPIX;
  float* ssimPart = msePart + nMse;
  float* sobPart  = ssimPart + nSsim;

  prep_y_mse<<<nMse, 256, 0, stream>>>(fused, vis, ir, mean_vis, std_vis,
                                       mean_ir, std_ir, Yf, Yv, Yi, msePart);
  ssim_wmma<<<nSsim, 128, 0, stream>>>(Yf, Yv, ssimPart);
  sobel_kernel<<<nSob, 256, 0, stream>>>(Yf, Yv, Yi, sobPart, nSob);
  finalize_kernel<<<1, 256, 0, stream>>>(msePart, nMse, ssimPart, nSsim,
                                         sobPart, nSob, (float*)d_out);
}